// ForgetMult_8151847928120
// MI455X (gfx1250) — compile-verified
//
#include <hip/hip_runtime.h>
#include <math.h>

// ForgetMult (QRNN) scan over T=64 for xi:(8,128,64,32,32) f32.
// Memory-bound: 384 MiB traffic -> ~16.5us at 23.3 TB/s on MI455X.
// Per-chain decay coefficients => WMMA structurally inapplicable; the CDNA5
// feature that pays here is the async global->LDS DMA path (ASYNCcnt),
// quad-buffered with prefetch depth 2 for per-wave MLP + safe LDS reuse.

typedef float v4f __attribute__((ext_vector_type(4)));
typedef int   v4i __attribute__((ext_vector_type(4)));

#define B_DIM 8
#define C_DIM 64
#define T_LEN 64
#define P_LEN 1024            // 32*32 spatial, contiguous innermost
#define SLAB  (T_LEN * P_LEN) // 65536 elements per (b,ch) slab

#define THREADS 128           // 4 waves; each thread owns 4 chains (float4)
#define BLOCKS_PER_BC 2       // 128*4*2 = 1024 = P_LEN
#define GRID (B_DIM * C_DIM * BLOCKS_PER_BC)

#define NBUF 4                // quad buffer: reuse distance = 2 iterations

#if defined(__gfx1250__) && __has_builtin(__builtin_amdgcn_global_load_async_to_lds_b128)
#define HAVE_ASYNC_LDS 1
#else
#define HAVE_ASYNC_LDS 0
#endif

#if defined(__gfx1250__)
#if __has_builtin(__builtin_amdgcn_s_wait_asynccnt)
#define WAIT_ASYNC(n) __builtin_amdgcn_s_wait_asynccnt(n)
#else
#define WAIT_ASYNC(n) asm volatile("s_wait_asynccnt " #n ::: "memory")
#endif
#else
#define WAIT_ASYNC(n)
#endif

// Builtin signature wants int4* in AS1 (global) / AS3 (LDS).
typedef __attribute__((address_space(1))) v4i* gv4i_p;
typedef __attribute__((address_space(3))) v4i* lv4i_p;
typedef __attribute__((address_space(3))) void lvoid;

// Global VA == generic VA, so inttoptr into AS1 is exact (also strips const).
#define GADDR(p) ((gv4i_p)(unsigned long long)(p))
// LDS needs a real addrspacecast (generic LDS ptr is aperture-based).
#define LADDR(p) ((lv4i_p)(lvoid*)(p))

__device__ __forceinline__ v4f sigmoid4(v4f v) {
    v4f r;
    r.x = 1.0f / (1.0f + expf(-v.x));
    r.y = 1.0f / (1.0f + expf(-v.y));
    r.z = 1.0f / (1.0f + expf(-v.z));
    r.w = 1.0f / (1.0f + expf(-v.w));
    return r;
}

__global__ __launch_bounds__(THREADS)
void forget_mult_scan(const float* __restrict__ xi, float* __restrict__ out) {
    const int tid  = threadIdx.x;
    const int bc   = blockIdx.x >> 1;          // b*64 + c
    const int half = blockIdx.x & 1;
    const int b = bc >> 6;
    const int c = bc & 63;
    const int p = half * (THREADS * 4) + tid * 4;   // [0,1024), 16B aligned

    const size_t fBase = (size_t)(b * 128 + c)      * SLAB + p;  // sigmoid gate half
    const size_t xBase = (size_t)(b * 128 + 64 + c) * SLAB + p;  // value half
    const size_t oBase = (size_t)(b * 64 + c)       * SLAB + p;

    const float* fG = xi + fBase;
    const float* xG = xi + xBase;
    float*       oG = out + oBase;

    v4f h = (v4f)0.0f;

#if HAVE_ASYNC_LDS
    // Each lane stages its own 16B f/x chunk per timestep; no cross-wave
    // sharing -> no workgroup barrier, only ASYNCcnt ordering.
    __shared__ v4f sF[NBUF][THREADS];
    __shared__ v4f sX[NBUF][THREADS];

    // Prologue: keep 2 timesteps in flight.
    __builtin_amdgcn_global_load_async_to_lds_b128(GADDR(fG),         LADDR(&sF[0][tid]), 0, 0);
    __builtin_amdgcn_global_load_async_to_lds_b128(GADDR(xG),         LADDR(&sX[0][tid]), 0, 0);
    __builtin_amdgcn_global_load_async_to_lds_b128(GADDR(fG + P_LEN), LADDR(&sF[1][tid]), 0, 0);
    __builtin_amdgcn_global_load_async_to_lds_b128(GADDR(xG + P_LEN), LADDR(&sX[1][tid]), 0, 0);

    #pragma unroll 4
    for (int t = 0; t < T_LEN; ++t) {
        const int cur = t & (NBUF - 1);
        if (t + 2 < T_LEN) {
            // Prefetch t+2 into a buffer last touched (read) at t-2; the
            // interleaving s_wait_asynccnt (global-latency scale) guarantees
            // that ds_load retired long before this DMA can write LDS.
            const int nxt = (t + 2) & (NBUF - 1);
            __builtin_amdgcn_global_load_async_to_lds_b128(
                GADDR(fG + (size_t)(t + 2) * P_LEN), LADDR(&sF[nxt][tid]), 0, 0);
            __builtin_amdgcn_global_load_async_to_lds_b128(
                GADDR(xG + (size_t)(t + 2) * P_LEN), LADDR(&sX[nxt][tid]), 0, 0);
            WAIT_ASYNC(4);   // pairs {t+1, t+2} may remain in flight
        } else if (t + 1 < T_LEN) {
            WAIT_ASYNC(2);   // pair {t+1} may remain in flight
        } else {
            WAIT_ASYNC(0);
        }
        v4f fr = sF[cur][tid];   // ds_load_b128
        v4f xv = sX[cur][tid];
        v4f f  = sigmoid4(fr);
        h = f * xv + (1.0f - f) * h;   // reference formulation; contracts to FMAs
        __builtin_nontemporal_store(h, (v4f*)(oG + (size_t)t * P_LEN));
    }
#else
    // Fallback: plain vectorized streaming loop (compiler software-pipelines
    // the independent b128 loads; scoreboarded via LOADcnt).
    const v4f* fV = (const v4f*)fG;
    const v4f* xV = (const v4f*)xG;
    #pragma unroll 4
    for (int t = 0; t < T_LEN; ++t) {
        v4f fr = __builtin_nontemporal_load(fV + (size_t)t * (P_LEN / 4));
        v4f xv = __builtin_nontemporal_load(xV + (size_t)t * (P_LEN / 4));
        v4f f  = sigmoid4(fr);
        h = f * xv + (1.0f - f) * h;
        __builtin_nontemporal_store(h, (v4f*)(oG + (size_t)t * P_LEN));
    }
#endif
}

extern "C" void kernel_launch(void* const* d_in, const int* in_sizes, int n_in,
                              void* d_out, int out_size, void* d_ws, size_t ws_size,
                              hipStream_t stream) {
    (void)in_sizes; (void)n_in; (void)out_size; (void)d_ws; (void)ws_size;
    const float* xi = (const float*)d_in[0];
    float* out = (float*)d_out;
    hipLaunchKernelGGL(forget_mult_scan, dim3(GRID), dim3(THREADS), 0, stream, xi, out);
}